// Classifier_49220325212327
// MI455X (gfx1250) — compile-verified
//
#include <hip/hip_runtime.h>

#define Nn 100000
#define Ee 1600000
#define Gg 64
#define Dd 128
#define Cc 10

typedef __attribute__((ext_vector_type(2))) float v2f;
typedef __attribute__((ext_vector_type(8))) float v8f;

// ---------------- zero ----------------
__global__ __launch_bounds__(256) void k_zero(float* __restrict__ p, int n4) {
  int i = blockIdx.x * 256 + threadIdx.x;
  if (i < n4) ((float4*)p)[i] = make_float4(0.f, 0.f, 0.f, 0.f);
}

// ---------------- degrees ----------------
__global__ __launch_bounds__(256) void k_degrees(const int* __restrict__ src,
                                                 const int* __restrict__ dst,
                                                 float* __restrict__ deg_out,
                                                 float* __restrict__ deg_in) {
  int e = blockIdx.x * 256 + threadIdx.x;
  if (e < Ee) {
    unsafeAtomicAdd(&deg_out[src[e]], 1.f);
    unsafeAtomicAdd(&deg_in[dst[e]], 1.f);
  }
}

__global__ __launch_bounds__(256) void k_norms(float* __restrict__ deg_out,
                                               float* __restrict__ deg_in) {
  int i = blockIdx.x * 256 + threadIdx.x;
  if (i < Nn) {
    deg_out[i] = rsqrtf(fmaxf(deg_out[i], 1.f));
    deg_in[i]  = rsqrtf(fmaxf(deg_in[i], 1.f));
  }
}

// ---------------- GEMM: T = (H * norm_out[:,None]) @ W,  fp32 WMMA ----------------
// Block: 256 threads = 8 waves. Block owns 16 rows; wave w owns cols [16w,16w+16).
// K loop: 128 / 4 = 32 x v_wmma_f32_16x16x4_f32 per wave.
__global__ __launch_bounds__(256) void k_gemm(const float* __restrict__ H,
                                              const float* __restrict__ W,
                                              const float* __restrict__ norm_out,
                                              float* __restrict__ T) {
  __shared__ __align__(16) float sW[Dd * Dd];   // 64 KB
  __shared__ __align__(16) float sA[16 * Dd];   // 8 KB
  const int tid  = threadIdx.x;
  const int row0 = blockIdx.x * 16;             // N = 6250 * 16, exact

  // Stage W into LDS (16 float4 per thread)
  for (int i = tid * 4; i < Dd * Dd; i += 256 * 4)
    *(float4*)&sW[i] = *(const float4*)&W[i];

  // Stage scaled A-tile into LDS (2 float4 per thread)
  for (int i = tid * 4; i < 16 * Dd; i += 256 * 4) {
    int r = i >> 7, c = i & (Dd - 1);
    float s = norm_out[row0 + r];
    float4 v = *(const float4*)&H[(size_t)(row0 + r) * Dd + c];
    v.x *= s; v.y *= s; v.z *= s; v.w *= s;
    *(float4*)&sA[i] = v;
  }
  __syncthreads();

  const int lane = tid & 31;
  const int wv   = tid >> 5;            // column tile 0..7
  const int l16  = lane & 15;           // M (for A) / N (for B,C)
  const int kh   = (lane >> 4) << 1;    // lanes 0-15 -> K={k,k+1}; 16-31 -> {k+2,k+3}
  const int col  = wv * 16 + l16;

  v8f acc = {};
  #pragma unroll
  for (int k = 0; k < Dd; k += 4) {
    v2f a = *(const v2f*)&sA[l16 * Dd + k + kh];          // A[M][k+kh], A[M][k+kh+1]
    v2f b;
    b.x = sW[(k + kh) * Dd + col];                        // B[k+kh][col]
    b.y = sW[(k + kh + 1) * Dd + col];                    // B[k+kh+1][col]
    acc = __builtin_amdgcn_wmma_f32_16x16x4_f32(
        /*neg_a=*/false, a, /*neg_b=*/false, b,
        /*c_mod=*/(short)0, acc, /*reuse_a=*/false, /*reuse_b=*/false);
  }

  // C/D layout: VGPR r -> M=r (lanes 0-15), M=r+8 (lanes 16-31); N = l16
  const int mbase = row0 + ((lane < 16) ? 0 : 8);
  #pragma unroll
  for (int r = 0; r < 8; ++r)
    T[(size_t)(mbase + r) * Dd + col] = acc[r];
}

// ---------------- SpMM scatter: M[dst] += T[src] ----------------
// One wave per edge; 4 floats per lane; hardware f32 atomics.
__global__ __launch_bounds__(256) void k_scatter(const float* __restrict__ T,
                                                 const int* __restrict__ src,
                                                 const int* __restrict__ dst,
                                                 float* __restrict__ M) {
  int gt = blockIdx.x * 256 + threadIdx.x;
  int e = gt >> 5;
  if (e >= Ee) return;
  int lane = gt & 31;
  int s = src[e], d = dst[e];
  float4 v = *(const float4*)&T[(size_t)s * Dd + lane * 4];
  float* o = &M[(size_t)d * Dd + lane * 4];
  unsafeAtomicAdd(o + 0, v.x);
  unsafeAtomicAdd(o + 1, v.y);
  unsafeAtomicAdd(o + 2, v.z);
  unsafeAtomicAdd(o + 3, v.w);
}

// ---------------- post: h = relu(m * norm_in + b), in place ----------------
__global__ __launch_bounds__(256) void k_post(float* __restrict__ M,
                                              const float* __restrict__ norm_in,
                                              const float* __restrict__ b) {
  int gt = blockIdx.x * 256 + threadIdx.x;
  int node = gt >> 5;
  if (node >= Nn) return;
  int c = (gt & 31) * 4;
  float s = norm_in[node];
  float4 v  = *(const float4*)&M[(size_t)node * Dd + c];
  float4 bb = *(const float4*)&b[c];
  v.x = fmaxf(fmaf(v.x, s, bb.x), 0.f);
  v.y = fmaxf(fmaf(v.y, s, bb.y), 0.f);
  v.z = fmaxf(fmaf(v.z, s, bb.z), 0.f);
  v.w = fmaxf(fmaf(v.w, s, bb.w), 0.f);
  *(float4*)&M[(size_t)node * Dd + c] = v;
}

// ---------------- per-graph pooling (sums + counts) ----------------
__global__ __launch_bounds__(256) void k_pool(const float* __restrict__ H,
                                              const int* __restrict__ gid,
                                              float* __restrict__ pool,
                                              float* __restrict__ cnt) {
  int gt = blockIdx.x * 256 + threadIdx.x;
  int node = gt >> 5;
  if (node >= Nn) return;
  int c = (gt & 31) * 4;
  int g = gid[node];
  float4 v = *(const float4*)&H[(size_t)node * Dd + c];
  float* o = &pool[g * Dd + c];
  unsafeAtomicAdd(o + 0, v.x);
  unsafeAtomicAdd(o + 1, v.y);
  unsafeAtomicAdd(o + 2, v.z);
  unsafeAtomicAdd(o + 3, v.w);
  if ((gt & 31) == 0) unsafeAtomicAdd(&cnt[g], 1.f);
}

// ---------------- classifier: hg = pool/cnt; logits = hg @ Wc + bc ----------------
__global__ __launch_bounds__(128) void k_classifier(const float* __restrict__ pool,
                                                    const float* __restrict__ cnt,
                                                    const float* __restrict__ Wc,
                                                    const float* __restrict__ bc,
                                                    float* __restrict__ out) {
  __shared__ float sh[Dd];
  int g = blockIdx.x, t = threadIdx.x;
  float inv = 1.f / fmaxf(cnt[g], 1.f);
  float hv = pool[g * Dd + t] * inv;
  out[Gg * Cc + g * Dd + t] = hv;   // hg, after logits block
  sh[t] = hv;
  __syncthreads();
  if (t < Cc) {
    float a = bc[t];
    #pragma unroll 8
    for (int k = 0; k < Dd; ++k) a = fmaf(sh[k], Wc[k * Cc + t], a);
    out[g * Cc + t] = a;            // logits
  }
}

extern "C" void kernel_launch(void* const* d_in, const int* in_sizes, int n_in,
                              void* d_out, int out_size, void* d_ws, size_t ws_size,
                              hipStream_t stream) {
  (void)in_sizes; (void)n_in; (void)out_size; (void)ws_size;
  const float* emb = (const float*)d_in[0];
  const int*   src = (const int*)d_in[1];
  const int*   dst = (const int*)d_in[2];
  const int*   gid = (const int*)d_in[3];
  const float* Wl[4] = {(const float*)d_in[4], (const float*)d_in[6],
                        (const float*)d_in[8], (const float*)d_in[10]};
  const float* bl[4] = {(const float*)d_in[5], (const float*)d_in[7],
                        (const float*)d_in[9], (const float*)d_in[11]};
  const float* Wc = (const float*)d_in[12];
  const float* bc = (const float*)d_in[13];
  float* out = (float*)d_out;

  // Workspace layout (floats): [norm_out N][norm_in N][X N*D][Y N*D][pool G*D][cnt G]
  float* ws       = (float*)d_ws;
  float* norm_out = ws;
  float* norm_in  = ws + Nn;
  float* X        = ws + 2 * (size_t)Nn;
  float* Y        = X + (size_t)Nn * Dd;
  float* pool     = Y + (size_t)Nn * Dd;
  float* cnt      = pool + Gg * Dd;

  auto zero = [&](float* p, size_t n) {
    int n4 = (int)(n / 4);
    k_zero<<<(n4 + 255) / 256, 256, 0, stream>>>(p, n4);
  };

  // degrees -> norms
  zero(norm_out, 2 * (size_t)Nn);
  k_degrees<<<(Ee + 255) / 256, 256, 0, stream>>>(src, dst, norm_out, norm_in);
  k_norms<<<(Nn + 255) / 256, 256, 0, stream>>>(norm_out, norm_in);

  // 4 GraphConv layers (ping-pong: gemm -> Y, scatter -> X, post in place)
  const float* hin = emb;
  for (int l = 0; l < 4; ++l) {
    k_gemm<<<Nn / 16, 256, 0, stream>>>(hin, Wl[l], norm_out, Y);
    zero(X, (size_t)Nn * Dd);
    k_scatter<<<((size_t)Ee * 32 + 255) / 256, 256, 0, stream>>>(Y, src, dst, X);
    k_post<<<(Nn * 32 + 255) / 256, 256, 0, stream>>>(X, norm_in, bl[l]);
    hin = X;
  }

  // pooling + classifier
  zero(pool, (size_t)Gg * Dd + Gg);
  k_pool<<<(Nn * 32 + 255) / 256, 256, 0, stream>>>(X, gid, pool, cnt);
  k_classifier<<<Gg, 128, 0, stream>>>(pool, cnt, Wc, bc, out);
}